// HyperbolicGraphConvolution_89653147337382
// MI455X (gfx1250) — compile-verified
//
#include <hip/hip_runtime.h>
#include <math.h>

typedef __attribute__((ext_vector_type(2))) float v2f;
typedef __attribute__((ext_vector_type(8))) float v8f;

#define FDIM 128
#define MIN_NORM 1e-15f
#define MAXN 0.996f        /* (1 - 4e-3) / sqrt(c), c = 1 */
#define ATC  0.99999990f   /* 1 - 1e-7 artanh clip */
#define MAXV 1e6f

__device__ __forceinline__ float wsum(float v) {
#pragma unroll
  for (int off = 16; off > 0; off >>= 1) v += __shfl_xor(v, off, 32);
  return v;
}

__device__ __forceinline__ float artanh_c(float x) {
  return atanhf(fminf(fmaxf(x, -ATC), ATC));
}

// ---------------------------------------------------------------------------
// Zero the segment-sum accumulator.
// ---------------------------------------------------------------------------
__global__ __launch_bounds__(256) void zero_kernel(float4* __restrict__ p, long long n4) {
  long long i = (long long)blockIdx.x * blockDim.x + threadIdx.x;
  if (i < n4) p[i] = make_float4(0.f, 0.f, 0.f, 0.f);
}

// ---------------------------------------------------------------------------
// mx = x @ W^T via V_WMMA_F32_16X16X4_F32. One wave per 16x16 output tile.
// A (16x4 f32, 2 VGPR): lane = M (lane&15), VGPR pair holds K-pair (lane>>4)*2.
// B (4x16 f32, 2 VGPR): lane = N, same K-pair split. B[k][n] = W[n][k], so the
// B fragment is read straight out of row n of W (row-major) — no transpose.
// ---------------------------------------------------------------------------
__global__ __launch_bounds__(256) void gemm_wmma_kernel(const float* __restrict__ X,
                                                        const float* __restrict__ W,
                                                        float* __restrict__ MX,
                                                        int n_row_tiles) {
  const int wave = (int)((blockIdx.x * blockDim.x + threadIdx.x) >> 5);
  const int lane = threadIdx.x & 31;
  const int col_tile = wave & 7;   // 128/16 = 8 column tiles
  const int row_tile = wave >> 3;
  if (row_tile >= n_row_tiles) return;   // wave-uniform

  const int mn = lane & 15;
  const int kk = (lane >> 4) << 1;       // 0 or 2
  const float* __restrict__ xrow = X + (size_t)(row_tile * 16 + mn) * FDIM;
  const float* __restrict__ wrow = W + (size_t)(col_tile * 16 + mn) * FDIM;

  v8f c = {};
#pragma unroll 4
  for (int k = 0; k < FDIM; k += 4) {
    v2f a = *(const v2f*)(xrow + k + kk);
    v2f b = *(const v2f*)(wrow + k + kk);
    c = __builtin_amdgcn_wmma_f32_16x16x4_f32(false, a, false, b, (short)0, c,
                                              false, false);
  }

  // C/D layout: lanes 0-15 -> M = g, lanes 16-31 -> M = g + 8; N = lane & 15.
  const int mbase = row_tile * 16 + ((lane >> 4) << 3);
  float* __restrict__ out = MX + (size_t)mbase * FDIM + col_tile * 16 + (lane & 15);
#pragma unroll
  for (int g = 0; g < 8; ++g) out[(size_t)g * FDIM] = c[g];
}

// ---------------------------------------------------------------------------
// Per-node: mobius_matvec scaling + proj + mobius_add(bias) + proj + logmap0.
// One wave32 per node; 32 lanes x float4 = 128 features.
// ---------------------------------------------------------------------------
__global__ __launch_bounds__(256) void tangent_kernel(const float* __restrict__ X,
                                                      const float* __restrict__ MX,
                                                      const float* __restrict__ B,
                                                      float* __restrict__ XT,
                                                      int nnode) {
  const int wave = (int)((blockIdx.x * blockDim.x + threadIdx.x) >> 5);
  const int lane = threadIdx.x & 31;
  if (wave >= nnode) return;   // wave-uniform

  const float4 x4 = *(const float4*)(X + (size_t)wave * FDIM + lane * 4);
  const float4 m4 = *(const float4*)(MX + (size_t)wave * FDIM + lane * 4);
  const float4 b4 = *(const float4*)(B + lane * 4);
  float xv[4] = {x4.x, x4.y, x4.z, x4.w};
  float mv[4] = {m4.x, m4.y, m4.z, m4.w};
  float bv[4] = {b4.x, b4.y, b4.z, b4.w};

  float xn2 = wsum(xv[0]*xv[0] + xv[1]*xv[1] + xv[2]*xv[2] + xv[3]*xv[3]);
  float mn2 = wsum(mv[0]*mv[0] + mv[1]*mv[1] + mv[2]*mv[2] + mv[3]*mv[3]);
  const float xn = fmaxf(sqrtf(xn2), MIN_NORM);
  const float mn = fmaxf(sqrtf(mn2), MIN_NORM);

  // mobius_matvec: tanh(|mx|/|x| * artanh(|x|)) * mx / |mx|
  float s = tanhf((mn / xn) * artanh_c(xn)) / mn;
  if (mn2 == 0.f) s = 0.f;
  float h[4];
#pragma unroll
  for (int i = 0; i < 4; ++i) h[i] = s * mv[i];
  float hn = fabsf(s) * sqrtf(mn2);
  if (hn > MAXN) {
    const float r = MAXN / hn;
#pragma unroll
    for (int i = 0; i < 4; ++i) h[i] *= r;
    hn = MAXN;
  }

  // hyp_bias = proj(expmap0(bias))
  const float bn2 = wsum(bv[0]*bv[0] + bv[1]*bv[1] + bv[2]*bv[2] + bv[3]*bv[3]);
  const float bn = fmaxf(sqrtf(bn2), MIN_NORM);
  const float tb = tanhf(bn);
  float sb = tb / bn;
  float hbn = tb;
  if (hbn > MAXN) { sb *= MAXN / hbn; hbn = MAXN; }
  float hb[4];
#pragma unroll
  for (int i = 0; i < 4; ++i) hb[i] = sb * bv[i];

  // mobius_add(h, hb)
  const float xy = wsum(h[0]*hb[0] + h[1]*hb[1] + h[2]*hb[2] + h[3]*hb[3]);
  const float x2 = hn * hn, y2 = hbn * hbn;
  const float ca = 1.f + 2.f * xy + y2;
  const float cb = 1.f - x2;
  const float den = fmaxf(1.f + 2.f * xy + x2 * y2, MIN_NORM);
  float r[4];
#pragma unroll
  for (int i = 0; i < 4; ++i) r[i] = (ca * h[i] + cb * hb[i]) / den;

  // proj
  const float rn2 = wsum(r[0]*r[0] + r[1]*r[1] + r[2]*r[2] + r[3]*r[3]);
  float rn = sqrtf(rn2);
  if (rn > MAXN) {
    const float sc = MAXN / rn;
#pragma unroll
    for (int i = 0; i < 4; ++i) r[i] *= sc;
    rn = MAXN;
  }

  // logmap0
  const float nn = fmaxf(rn, MIN_NORM);
  const float u = artanh_c(nn) / nn;
  float4 o;
  o.x = u * r[0]; o.y = u * r[1]; o.z = u * r[2]; o.w = u * r[3];
  *(float4*)(XT + (size_t)wave * FDIM + lane * 4) = o;
}

// ---------------------------------------------------------------------------
// Edge scatter: support[row] += adj_val * x_tangent[col]. One wave per edge.
// Atomic FP32 adds -> global_atomic_add_f32 (relaxed, agent scope); the
// 25.6 MB accumulator is L2-resident (192 MB L2) so atomics stay on-chip.
// ---------------------------------------------------------------------------
__global__ __launch_bounds__(256) void scatter_kernel(const float* __restrict__ XT,
                                                      const int* __restrict__ EROW,
                                                      const int* __restrict__ ECOL,
                                                      const float* __restrict__ AV,
                                                      float* __restrict__ S,
                                                      int ne) {
  const int wave = (int)((blockIdx.x * blockDim.x + threadIdx.x) >> 5);
  const int lane = threadIdx.x & 31;
  if (wave >= ne) return;   // wave-uniform

  const int row = EROW[wave];
  const int col = ECOL[wave];
  const float a = AV[wave];
  const float4 t = *(const float4*)(XT + (size_t)col * FDIM + lane * 4);
  float* __restrict__ dst = S + (size_t)row * FDIM + lane * 4;
  __hip_atomic_fetch_add(dst + 0, a * t.x, __ATOMIC_RELAXED, __HIP_MEMORY_SCOPE_AGENT);
  __hip_atomic_fetch_add(dst + 1, a * t.y, __ATOMIC_RELAXED, __HIP_MEMORY_SCOPE_AGENT);
  __hip_atomic_fetch_add(dst + 2, a * t.z, __ATOMIC_RELAXED, __HIP_MEMORY_SCOPE_AGENT);
  __hip_atomic_fetch_add(dst + 3, a * t.w, __ATOMIC_RELAXED, __HIP_MEMORY_SCOPE_AGENT);
}

// ---------------------------------------------------------------------------
// Per-node: clamp + proj(expmap0(support)) + relu(logmap0) clamp + expmap0.
// ---------------------------------------------------------------------------
__global__ __launch_bounds__(256) void finish_kernel(const float* __restrict__ S,
                                                     float* __restrict__ O,
                                                     int nnode) {
  const int wave = (int)((blockIdx.x * blockDim.x + threadIdx.x) >> 5);
  const int lane = threadIdx.x & 31;
  if (wave >= nnode) return;   // wave-uniform

  const float4 s4 = *(const float4*)(S + (size_t)wave * FDIM + lane * 4);
  float s[4] = {fminf(s4.x, MAXV), fminf(s4.y, MAXV), fminf(s4.z, MAXV), fminf(s4.w, MAXV)};

  // h = proj(expmap0(support))
  const float n2 = wsum(s[0]*s[0] + s[1]*s[1] + s[2]*s[2] + s[3]*s[3]);
  const float n = fmaxf(sqrtf(n2), MIN_NORM);
  const float th = tanhf(n);
  float sc = th / n;
  float hn = th;
  float h[4];
#pragma unroll
  for (int i = 0; i < 4; ++i) h[i] = sc * s[i];
  if (hn > MAXN) {
    const float r = MAXN / hn;
#pragma unroll
    for (int i = 0; i < 4; ++i) h[i] *= r;
    hn = MAXN;
  }

  // xt = min(relu(logmap0(h)), MAXV)
  const float nn = fmaxf(hn, MIN_NORM);
  const float u = artanh_c(nn) / nn;
  float t[4];
#pragma unroll
  for (int i = 0; i < 4; ++i) t[i] = fminf(fmaxf(u * h[i], 0.f), MAXV);

  // out = expmap0(xt)   (no final proj in the reference)
  const float tn2 = wsum(t[0]*t[0] + t[1]*t[1] + t[2]*t[2] + t[3]*t[3]);
  const float tn = fmaxf(sqrtf(tn2), MIN_NORM);
  const float oc = tanhf(tn) / tn;
  float4 o;
  o.x = oc * t[0]; o.y = oc * t[1]; o.z = oc * t[2]; o.w = oc * t[3];
  *(float4*)(O + (size_t)wave * FDIM + lane * 4) = o;
}

// ---------------------------------------------------------------------------
extern "C" void kernel_launch(void* const* d_in, const int* in_sizes, int n_in,
                              void* d_out, int out_size, void* d_ws, size_t ws_size,
                              hipStream_t stream) {
  const float* x    = (const float*)d_in[0];
  const float* adj  = (const float*)d_in[1];
  const float* w    = (const float*)d_in[2];
  const float* bias = (const float*)d_in[3];
  const int*   erow = (const int*)d_in[4];
  const int*   ecol = (const int*)d_in[5];

  const int N = in_sizes[0] / FDIM;   // 50000
  const int E = in_sizes[1];          // 800000

  float* mx  = (float*)d_ws;                 // [N,128]
  float* xt  = mx + (size_t)N * FDIM;        // [N,128]
  float* sup = xt + (size_t)N * FDIM;        // [N,128]
  float* out = (float*)d_out;

  // 1) zero segment-sum accumulator
  const long long n4 = (long long)N * (FDIM / 4);
  zero_kernel<<<(int)((n4 + 255) / 256), 256, 0, stream>>>((float4*)sup, n4);

  // 2) mx = x @ W^T  (WMMA fp32; one block = one 16-row tile x 8 col tiles)
  const int row_tiles = N / 16;   // N is a multiple of 16
  gemm_wmma_kernel<<<row_tiles, 256, 0, stream>>>(x, w, mx, row_tiles);

  // 3) HypLinear tail + logmap0 -> x_tangent
  tangent_kernel<<<(N + 7) / 8, 256, 0, stream>>>(x, mx, bias, xt, N);

  // 4) edge gather/scale/scatter-add (memory-bound phase, L2-resident)
  scatter_kernel<<<(E + 7) / 8, 256, 0, stream>>>(xt, erow, ecol, adj, sup, E);

  // 5) clamp + expmap0/proj + relu-in-tangent + expmap0 -> out
  finish_kernel<<<(N + 7) / 8, 256, 0, stream>>>(sup, out, N);
}